// PNABlock_70300024701669
// MI455X (gfx1250) — compile-verified
//
#include <hip/hip_runtime.h>
#include <hip/hip_bf16.h>
#include <float.h>
#include <math.h>

#define N_NODES 50000
#define N_EDGES 800000
#define FIN 64
#define TD 256          // T * F_IN
#define ODIM 64         // T * F_OUT
#define FPOST 832       // 13 * F_IN
#define EPS_STD 1e-5f
#define EPS_BN 1e-5f

typedef float v2f __attribute__((ext_vector_type(2)));
typedef float v8f __attribute__((ext_vector_type(8)));

// ---------------- utility fills ----------------
__global__ void fill_kernel(float* __restrict__ p, float v, long n) {
    long i = (long)blockIdx.x * blockDim.x + threadIdx.x;
    long st = (long)gridDim.x * blockDim.x;
    for (; i < n; i += st) p[i] = v;
}

// transpose W_post (64 x 832) -> WpT (832 x 64), W_lin (64 x 64) -> WlT (64 x 64)
__global__ void prep_weights(const float* __restrict__ Wpost, const float* __restrict__ Wlin,
                             float* __restrict__ WpT, float* __restrict__ WlT) {
    int st = gridDim.x * blockDim.x;
    for (int idx = blockIdx.x * blockDim.x + threadIdx.x; idx < FPOST * ODIM; idx += st) {
        int i = idx >> 6, j = idx & 63;
        WpT[idx] = Wpost[j * FPOST + i];
    }
    for (int idx = blockIdx.x * blockDim.x + threadIdx.x; idx < ODIM * ODIM; idx += st) {
        int k = idx >> 6, j = idx & 63;
        WlT[idx] = Wlin[j * ODIM + k];
    }
}

// ---------------- WMMA f32 GEMM: C[N x P] = X[N x 64] * B (+bias) ----------------
// B[k, col] = Wbase[col*ldw + woff + k]
__global__ void gemm_x64_wmma(const float* __restrict__ X,
                              const float* __restrict__ Wbase, int ldw, int woff,
                              const float* __restrict__ bias,
                              float* __restrict__ C, int P) {
    const int wavesPerBlock = blockDim.x >> 5;
    int wave = blockIdx.x * wavesPerBlock + (threadIdx.x >> 5);
    int tilesPerRow = P >> 4;
    int totalTiles = (N_NODES >> 4) * tilesPerRow;
    if (wave >= totalTiles) return;
    int rt = wave / tilesPerRow, ct = wave % tilesPerRow;
    int row0 = rt << 4, col0 = ct << 4;
    int lane = threadIdx.x & 31;
    int hi   = lane >> 4;     // 0: K pair {0,1}, 1: K pair {2,3}
    int l15  = lane & 15;

    v8f c = {};
    const float* xrow = X + (long)(row0 + l15) * FIN;
    const float* wrow = Wbase + (long)(col0 + l15) * ldw + woff;
#pragma unroll
    for (int k = 0; k < FIN; k += 4) {
        int ka = k + hi * 2;
        v2f a, b;
        a.x = xrow[ka];     a.y = xrow[ka + 1];
        b.x = wrow[ka];     b.y = wrow[ka + 1];
        c = __builtin_amdgcn_wmma_f32_16x16x4_f32(false, a, false, b, (short)0, c,
                                                  false, false);
    }
#pragma unroll
    for (int r = 0; r < 8; ++r) {
        int m = row0 + r + hi * 8;
        int n = col0 + l15;
        float v = c[r];
        if (bias) v += bias[n];
        C[(long)m * P + n] = v;
    }
}

// ---------------- edge aggregation (atomics into L2-resident buffers) ----------------
__device__ __forceinline__ void atomicMaxF(float* p, float v) {
    if (v >= 0.0f) atomicMax((int*)p, __float_as_int(v));
    else           atomicMin((unsigned int*)p, __float_as_uint(v));
}
__device__ __forceinline__ void atomicMinF(float* p, float v) {
    if (v >= 0.0f) atomicMin((int*)p, __float_as_int(v));
    else           atomicMax((unsigned int*)p, __float_as_uint(v));
}

__global__ void edge_kernel(const int* __restrict__ ei, const float* __restrict__ x,
                            const float* __restrict__ u,
                            float* __restrict__ Ax, float* __restrict__ Aq,
                            float* __restrict__ mx, float* __restrict__ mn,
                            float* __restrict__ cnt) {
    int lane = threadIdx.x & 31;
    int wave = blockIdx.x * (blockDim.x >> 5) + (threadIdx.x >> 5);
    int nw = gridDim.x * (blockDim.x >> 5);
    for (int e = wave; e < N_EDGES; e += nw) {
        int s = ei[e];
        int d = ei[N_EDGES + e];
        if (lane == 0) atomicAdd(&cnt[d], 1.0f);
#pragma unroll
        for (int i = 0; i < 2; ++i) {
            int f = i * 32 + lane;
            atomicAdd(&Ax[(long)d * FIN + f], x[(long)s * FIN + f]);
        }
#pragma unroll
        for (int i = 0; i < 8; ++i) {
            int j = i * 32 + lane;
            float v = u[(long)s * TD + j];
            atomicAdd(&Aq[(long)d * TD + j], v * v);
            atomicMaxF(&mx[(long)d * TD + j], v);
            atomicMinF(&mn[(long)d * TD + j], v);
        }
    }
}

// ---------------- sum of log(cnt+1) over nodes ----------------
__global__ void degsum_kernel(const float* __restrict__ cnt, float* __restrict__ sumlog) {
    __shared__ float red[256];
    float s = 0.0f;
    for (int i = blockIdx.x * blockDim.x + threadIdx.x; i < N_NODES;
         i += gridDim.x * blockDim.x)
        s += logf(cnt[i] + 1.0f);
    red[threadIdx.x] = s;
    __syncthreads();
    for (int k = blockDim.x >> 1; k > 0; k >>= 1) {
        if (threadIdx.x < k) red[threadIdx.x] += red[threadIdx.x + k];
        __syncthreads();
    }
    if (threadIdx.x == 0) atomicAdd(sumlog, red[0]);
}

// ---------------- per-node finalize: agg stats -> post GEMM -> lin -> BN partials ----
__global__ __launch_bounds__(64)
void node_post_kernel(const float* __restrict__ x, const float* __restrict__ a,
                      const float* __restrict__ Au, const float* __restrict__ Aq,
                      const float* __restrict__ mx, const float* __restrict__ mn,
                      const float* __restrict__ cnt, const float* __restrict__ sumlog,
                      const float* __restrict__ WpT, const float* __restrict__ bpost,
                      const float* __restrict__ WlT, const float* __restrict__ blin,
                      float* __restrict__ h2, float* __restrict__ bnsum,
                      float* __restrict__ bnsumsq) {
    __shared__ float xsh[FIN];
    __shared__ float aggsh[4 * 256];
    __shared__ float postsh[ODIM];
    int n = blockIdx.x;
    int j = threadIdx.x;            // 0..63
    float cv = cnt[n];
    float denom = fmaxf(cv, 1.0f);
    bool  has = (cv > 0.0f);
    float avg_log = sumlog[0] / (float)N_NODES;
    float log_d = logf(denom + 1.0f);
    float amp = log_d / avg_log;
    float att = avg_log / log_d;

    xsh[j] = x[(long)n * FIN + j];
#pragma unroll
    for (int t = 0; t < 4; ++t) {
        int idx = t * 64 + j;
        long o = (long)n * TD + idx;
        float av = a[o], Auv = Au[o], Aqv = Aq[o];
        float s1 = cv * av + Auv;
        float mean = s1 / denom;
        float s2 = cv * av * av + 2.0f * av * Auv + Aqv;
        float var = fmaxf(s2 / denom - mean * mean, 0.0f);
        float stdv = sqrtf(var + EPS_STD);
        float mxo = has ? (av + mx[o]) : 0.0f;
        float mno = has ? (av + mn[o]) : 0.0f;
        aggsh[t * 256 + j]        = mean;
        aggsh[t * 256 + 64 + j]   = mxo;
        aggsh[t * 256 + 128 + j]  = mno;
        aggsh[t * 256 + 192 + j]  = stdv;
    }
    __syncthreads();

    {   // post[j], j enumerates (t,o), t = j>>4, o = j&15 ; WpT[i*64+j] coalesced
        int t = j >> 4;
        float s = bpost[j];
        for (int i = 0; i < FIN; ++i) s += WpT[i * 64 + j] * xsh[i];
        const float* aggt = &aggsh[t * 256];
        for (int i = 0; i < 256; ++i) {
            float g = aggt[i];
            s += g * (WpT[(64 + i) * 64 + j] + amp * WpT[(320 + i) * 64 + j]
                      + att * WpT[(576 + i) * 64 + j]);
        }
        postsh[j] = s;
    }
    __syncthreads();
    {
        float s = blin[j];
        for (int k = 0; k < ODIM; ++k) s += WlT[k * 64 + j] * postsh[k];
        h2[(long)n * ODIM + j] = s;
        atomicAdd(&bnsum[j], s);
        atomicAdd(&bnsumsq[j], s * s);
    }
}

__global__ void bn_finalize(const float* __restrict__ bnsum, const float* __restrict__ bnsumsq,
                            float* __restrict__ mu, float* __restrict__ inv) {
    int j = threadIdx.x;
    float m = bnsum[j] / (float)N_NODES;
    float v = bnsumsq[j] / (float)N_NODES - m * m;
    mu[j] = m;
    inv[j] = rsqrtf(fmaxf(v, 0.0f) + EPS_BN);
}

__global__ void bn_apply(const float* __restrict__ h2, const float* __restrict__ mu,
                         const float* __restrict__ inv, const float* __restrict__ gamma,
                         const float* __restrict__ beta, float* __restrict__ out) {
    long total = (long)N_NODES * ODIM;
    for (long i = (long)blockIdx.x * blockDim.x + threadIdx.x; i < total;
         i += (long)gridDim.x * blockDim.x) {
        int j = (int)(i & 63);
        float v = (h2[i] - mu[j]) * inv[j] * gamma[j] + beta[j];
        out[i] = fmaxf(v, 0.0f);
    }
}

// ---------------- launch ----------------
extern "C" void kernel_launch(void* const* d_in, const int* in_sizes, int n_in,
                              void* d_out, int out_size, void* d_ws, size_t ws_size,
                              hipStream_t stream) {
    const float* x      = (const float*)d_in[0];
    const int*   ei     = (const int*)d_in[1];
    const float* W_pre  = (const float*)d_in[2];   // (4,64,128): [:, :, 0:64]=Wi, 64:128=Wj
    const float* b_pre  = (const float*)d_in[3];   // 256
    const float* W_post = (const float*)d_in[4];   // (64, 832) flat
    const float* b_post = (const float*)d_in[5];   // 64
    const float* W_lin  = (const float*)d_in[6];   // 64x64
    const float* b_lin  = (const float*)d_in[7];
    const float* gamma  = (const float*)d_in[8];
    const float* beta   = (const float*)d_in[9];
    float* out = (float*)d_out;
    float* ws = (float*)d_ws;

    const long Nl = N_NODES;
    long off_u     = 0;                       // N*256 : u = x@Wj^T, later reused for Au
    long off_a     = off_u + Nl * TD;         // N*256 : a = x@Wi^T + b_pre
    long off_Ax    = off_a + Nl * TD;         // N*64  : segment-sum of x[src]
    long off_Aq    = off_Ax + Nl * FIN;       // N*256 : segment-sum of u[src]^2
    long off_mx    = off_Aq + Nl * TD;        // N*256 : segment-max of u[src]
    long off_mn    = off_mx + Nl * TD;        // N*256 : segment-min of u[src]
    long off_cnt   = off_mn + Nl * TD;        // N
    long off_sml   = off_cnt + Nl;            // sumlog(1) bnsum(64) bnsumsq(64) mu(64) inv(64)
    long off_WpT   = off_sml + 257;           // 832*64
    long off_WlT   = off_WpT + (long)FPOST * ODIM; // 64*64
    long off_h2    = off_WlT + ODIM * ODIM;   // N*64

    float* u    = ws + off_u;
    float* aA   = ws + off_a;
    float* Ax   = ws + off_Ax;
    float* Aq   = ws + off_Aq;
    float* mx   = ws + off_mx;
    float* mn   = ws + off_mn;
    float* cnt  = ws + off_cnt;
    float* sumlog  = ws + off_sml;
    float* bnsum   = ws + off_sml + 1;
    float* bnsumsq = ws + off_sml + 65;
    float* mu      = ws + off_sml + 129;
    float* inv     = ws + off_sml + 193;
    float* WpT  = ws + off_WpT;
    float* WlT  = ws + off_WlT;
    float* h2   = ws + off_h2;

    // init accumulators
    fill_kernel<<<1024, 256, 0, stream>>>(Ax, 0.0f, Nl * (FIN + TD));      // Ax + Aq
    fill_kernel<<<1024, 256, 0, stream>>>(mx, -FLT_MAX, Nl * TD);
    fill_kernel<<<1024, 256, 0, stream>>>(mn,  FLT_MAX, Nl * TD);
    fill_kernel<<<512, 256, 0, stream>>>(cnt, 0.0f, Nl + 129);             // cnt + sumlog + bnsum + bnsumsq
    prep_weights<<<128, 256, 0, stream>>>(W_post, W_lin, WpT, WlT);

    // node-level WMMA GEMMs: u = x @ Wj^T ; a = x @ Wi^T + b_pre
    int tiles = (N_NODES / 16) * (TD / 16);            // 50000
    int gblocks = (tiles * 32 + 255) / 256;            // 8 waves/block
    gemm_x64_wmma<<<gblocks, 256, 0, stream>>>(x, W_pre, 2 * FIN, FIN, nullptr, u, TD);
    gemm_x64_wmma<<<gblocks, 256, 0, stream>>>(x, W_pre, 2 * FIN, 0, b_pre, aA, TD);

    // edge-level atomics (u + destinations are L2-resident on MI455X's 192MB L2)
    edge_kernel<<<2048, 256, 0, stream>>>(ei, x, u, Ax, Aq, mx, mn, cnt);

    degsum_kernel<<<256, 256, 0, stream>>>(cnt, sumlog);

    // Au = (A x) @ Wj^T  — overwrite u (dead after edge phase)
    gemm_x64_wmma<<<gblocks, 256, 0, stream>>>(Ax, W_pre, 2 * FIN, FIN, nullptr, u, TD);

    node_post_kernel<<<N_NODES, 64, 0, stream>>>(x, aA, u, Aq, mx, mn, cnt, sumlog,
                                                 WpT, b_post, WlT, b_lin,
                                                 h2, bnsum, bnsumsq);
    bn_finalize<<<1, 64, 0, stream>>>(bnsum, bnsumsq, mu, inv);
    bn_apply<<<2048, 256, 0, stream>>>(h2, mu, inv, gamma, beta, out);
}